// FeatureOctree_74749610820348
// MI455X (gfx1250) — compile-verified
//
#include <hip/hip_runtime.h>

// FeatureOctree gather + trilinear smoothstep interp for MI455X (gfx1250, wave32).
//
// Roofline: ~1.8 GB compulsory traffic (3 levels x 8 corners x 64B rows per
// point + indices + output) at 23.3 TB/s -> ~77us floor; ~1 GFLOP f32 ->
// purely gather-bound. WMMA does not apply (per-point B-matrix einsum), and
// TDM/async-to-LDS would add an LDS round trip for use-once data, so the
// optimal gfx1250 path is lane-cooperative b128 gathers:
//   - 4 lanes per point; each 64B feature row = one s_clause of 4x b128.
//   - all 24 index loads hoisted first, so level L+1 gathers issue while
//     level L's FMAs retire (single dependent-load step on the critical path).
//   - coalesced 512B/wave output stores.

__global__ __launch_bounds__(256) void FeatureOctree_74749610820348_kernel(
    const float* __restrict__ xin,
    const float* __restrict__ feat0,
    const float* __restrict__ feat1,
    const float* __restrict__ feat2,
    const long long* __restrict__ idx0,
    const long long* __restrict__ idx1,
    const long long* __restrict__ idx2,
    float* __restrict__ out,
    int n_pts)
{
    const int tid = blockIdx.x * blockDim.x + threadIdx.x;
    const int pt  = tid >> 2;   // point index (8 points per wave32)
    const int sub = tid & 3;    // which float4 slice of the 16-float row
    if (pt >= n_pts) return;

    // ---- Issue ALL address-producing loads up front -------------------------
    // Indices are int64 < 2M: load only the low dwords (8 x b32 per level).
    const unsigned* ip0 = (const unsigned*)(idx0 + (long long)pt * 8);
    const unsigned* ip1 = (const unsigned*)(idx1 + (long long)pt * 8);
    const unsigned* ip2 = (const unsigned*)(idx2 + (long long)pt * 8);

    unsigned j[24];
#pragma unroll
    for (int k = 0; k < 8; ++k) j[k]      = ip0[2 * k];
#pragma unroll
    for (int k = 0; k < 8; ++k) j[8 + k]  = ip1[2 * k];
#pragma unroll
    for (int k = 0; k < 8; ++k) j[16 + k] = ip2[2 * k];

    const float xx = xin[3 * pt + 0];
    const float xy = xin[3 * pt + 1];
    const float xz = xin[3 * pt + 2];

    // ---- Weight math (VALU) overlaps the outstanding index loads -----------
    // coords = 2^level * (x*0.5 + 0.5) = 2^(level-1) * (x+1); level = 12,11,10
    const float scales[3] = {2048.f, 1024.f, 512.f};
    float pw[24];
#pragma unroll
    for (int lv = 0; lv < 3; ++lv) {
        const float s  = scales[lv];
        const float cx = s * (xx + 1.f);
        const float cy = s * (xy + 1.f);
        const float cz = s * (xz + 1.f);
        const float dx = cx - floorf(cx);
        const float dy = cy - floorf(cy);
        const float dz = cz - floorf(cz);
        // smoothstep t = 3d^2 - 2d^3
        const float tx = dx * dx * (3.f - 2.f * dx);
        const float ty = dy * dy * (3.f - 2.f * dy);
        const float tz = dz * dz * (3.f - 2.f * dz);
        const float ux = 1.f - tx, uy = 1.f - ty, uz = 1.f - tz;

        const float pxy00 = ux * uy, pxy01 = ux * ty;
        const float pxy10 = tx * uy, pxy11 = tx * ty;
        float* p = pw + lv * 8;
        p[0] = pxy00 * uz; p[1] = pxy00 * tz;
        p[2] = pxy01 * uz; p[3] = pxy01 * tz;
        p[4] = pxy10 * uz; p[5] = pxy10 * tz;
        p[6] = pxy11 * uz; p[7] = pxy11 * tz;
    }

    // ---- Gather + accumulate, one clause of 8 b128 gathers per level -------
    const float* const feats[3] = {feat0, feat1, feat2};
    float4 acc = make_float4(0.f, 0.f, 0.f, 0.f);

#pragma unroll
    for (int lv = 0; lv < 3; ++lv) {
        const float* __restrict__ fb = feats[lv];
        float4 g[8];
#pragma unroll
        for (int k = 0; k < 8; ++k) {
            // 64B row gather: the 4 lanes of this point each fetch one
            // aligned float4 -> exactly one 64B segment per corner row.
            g[k] = ((const float4*)(fb + (size_t)j[lv * 8 + k] * 16))[sub];
        }
#pragma unroll
        for (int k = 0; k < 8; ++k) {
            const float w = pw[lv * 8 + k];
            acc.x = fmaf(w, g[k].x, acc.x);
            acc.y = fmaf(w, g[k].y, acc.y);
            acc.z = fmaf(w, g[k].z, acc.z);
            acc.w = fmaf(w, g[k].w, acc.w);
        }
    }

    // Coalesced: 8 consecutive points per wave -> contiguous 512B store burst.
    float4* op = (float4*)(out + (size_t)pt * 16);
    op[sub] = acc;
}

extern "C" void kernel_launch(void* const* d_in, const int* in_sizes, int n_in,
                              void* d_out, int out_size, void* d_ws, size_t ws_size,
                              hipStream_t stream)
{
    const float*     x  = (const float*)d_in[0];
    const float*     f0 = (const float*)d_in[1];
    const float*     f1 = (const float*)d_in[2];
    const float*     f2 = (const float*)d_in[3];
    const long long* i0 = (const long long*)d_in[4];
    const long long* i1 = (const long long*)d_in[5];
    const long long* i2 = (const long long*)d_in[6];
    float* out = (float*)d_out;

    const int n_pts   = in_sizes[0] / 3;       // x is (N_PTS, 3)
    const int threads = n_pts * 4;             // 4 lanes per point
    const int block   = 256;                   // 8 wave32s per block
    const int grid    = (threads + block - 1) / block;

    FeatureOctree_74749610820348_kernel<<<grid, block, 0, stream>>>(
        x, f0, f1, f2, i0, i1, i2, out, n_pts);
}